// ConditionalRandomField_43190191128876
// MI455X (gfx1250) — compile-verified
//
#include <hip/hip_runtime.h>

typedef float v2f __attribute__((ext_vector_type(2)));
typedef float v8f __attribute__((ext_vector_type(8)));

constexpr int BB = 128;   // batch
constexpr int SS = 1024;  // sequence length
constexpr int TT = 128;   // num tags
constexpr int BT = 16;    // batch rows per workgroup (WMMA M)
constexpr int APAD = 130; // padded LDS row stride (dwords) -> bank-conflict free

__global__ void crf_zero_out(float* out) { out[0] = 0.0f; }

// ---------------------------------------------------------------------------
// Numerator: per-batch gather score. One block per batch element.
// ---------------------------------------------------------------------------
__global__ __launch_bounds__(128) void crf_num_kernel(
    const float* __restrict__ inputs, const int* __restrict__ tags,
    const unsigned char* __restrict__ mask, const float* __restrict__ trans,
    const float* __restrict__ startt, const float* __restrict__ endt,
    float* __restrict__ out) {
  const int b = blockIdx.x;
  const int tid = threadIdx.x;
  const int* tg = tags + (size_t)b * SS;
  const unsigned char* mk = mask + (size_t)b * SS;
  const float* inp = inputs + (size_t)b * SS * TT;

  float psum = 0.0f;
  int pcnt = 0;
  for (int s = tid; s < SS; s += blockDim.x) {
    const float mf = mk[s] ? 1.0f : 0.0f;
    pcnt += mk[s] ? 1 : 0;
    if (s >= 1) psum += trans[tg[s - 1] * TT + tg[s]] * mf;        // transition score
    if (s <= SS - 2) psum += inp[(size_t)s * TT + tg[s]] * mf;     // emission score
  }

  __shared__ float rs[128];
  __shared__ int rc[128];
  rs[tid] = psum;
  rc[tid] = pcnt;
  __syncthreads();
  if (tid == 0) {
    float sc = startt[tg[0]];
    int cnt = 0;
    for (int i = 0; i < (int)blockDim.x; ++i) { sc += rs[i]; cnt += rc[i]; }
    int last = cnt - 1;
    if (last < 0) last = 0;
    const int lt = tg[last];
    const float lm = mk[SS - 1] ? 1.0f : 0.0f;
    sc += endt[lt] + inp[(size_t)(SS - 1) * TT + lt] * lm;
    atomicAdd(out, sc);  // numerator adds +score
  }
}

// ---------------------------------------------------------------------------
// Denominator: forward algorithm as per-step exp-space GEMM on f32 WMMA.
// One workgroup = 16 batch rows; 8 waves, wave w owns output columns
// [16w, 16w+16). E-matrix columns (exp(trans)) live in 64 VGPRs per wave
// for the entire scan; alpha lives in LDS.
// ---------------------------------------------------------------------------
__global__ __launch_bounds__(256) void crf_den_kernel(
    const float* __restrict__ inputs, const unsigned char* __restrict__ mask,
    const float* __restrict__ trans, const float* __restrict__ startt,
    const float* __restrict__ endt, float* __restrict__ out) {
  __shared__ float sh_alpha[BT * TT];    // alpha[m][j], stride 128
  __shared__ float sh_upd[BT * APAD];    // exp(alpha-m) then updates, stride 130
  __shared__ float sh_mrow[BT];          // per-row max

  const int tid = threadIdx.x;
  const int wave = tid >> 5;
  const int lane = tid & 31;
  const int subl = lane & 15;
  const int half = lane >> 4;
  const int b0 = blockIdx.x * BT;

  // --- Preload B fragments: E[k][j] = exp(trans[k][j]) for this wave's 16
  // columns. ISA B layout (4x16 f32): lanes 0-15 hold K = {0,1} in v0/v1,
  // lanes 16-31 hold K = {2,3}.
  v2f bfrag[32];
  {
    const int j = wave * 16 + subl;
#pragma unroll
    for (int c = 0; c < 32; ++c) {
      const int k0 = 4 * c + 2 * half;
      bfrag[c].x = __expf(trans[k0 * TT + j]);
      bfrag[c].y = __expf(trans[(k0 + 1) * TT + j]);
    }
  }

  // --- alpha0 = start_transitions + inputs[:, 0]
#pragma unroll
  for (int u = 0; u < (BT * TT) / 256; ++u) {
    const int e = tid + 256 * u;
    const int m = e >> 7, j = e & 127;
    sh_alpha[m * TT + j] = startt[j] + inputs[(size_t)(b0 + m) * SS * TT + j];
  }
  __syncthreads();

  for (int s = 1; s < SS; ++s) {
    // Phase A: per-row max. Wave w reduces rows 2w (lanes 0-15) / 2w+1.
    {
      const int row = 2 * wave + half;
      float mx = -3.4e38f;
      for (int c = subl; c < TT; c += 16) mx = fmaxf(mx, sh_alpha[row * TT + c]);
      mx = fmaxf(mx, __shfl_xor(mx, 1));
      mx = fmaxf(mx, __shfl_xor(mx, 2));
      mx = fmaxf(mx, __shfl_xor(mx, 4));
      mx = fmaxf(mx, __shfl_xor(mx, 8));
      if (subl == 0) sh_mrow[row] = mx;
    }
    __syncthreads();

    // Phase B: a[m][k] = exp(alpha[m][k] - mrow[m])
#pragma unroll
    for (int u = 0; u < (BT * TT) / 256; ++u) {
      const int e = tid + 256 * u;
      const int m = e >> 7, j = e & 127;
      sh_upd[m * APAD + j] = __expf(sh_alpha[m * TT + j] - sh_mrow[m]);
    }
    __syncthreads();

    // Phase C: C = A x E over K=128 with 32x V_WMMA_F32_16X16X4_F32.
    // A layout: lanes 0-15 row M=lane K={0,1}; lanes 16-31 row M=lane-16 K={2,3}.
    v8f acc = {};
    {
      const float* arow = &sh_upd[subl * APAD + 2 * half];
#pragma unroll
      for (int c = 0; c < 32; ++c) {
        v2f a;
        a.x = arow[4 * c];
        a.y = arow[4 * c + 1];
        acc = __builtin_amdgcn_wmma_f32_16x16x4_f32(
            false, a, false, bfrag[c], (short)0, acc, false, false);
      }
    }
    __syncthreads();  // all waves done reading sh_upd before we overwrite it

    // Phase C2: upd[m][j] = mrow[m] + log(acc).  C/D layout: VGPR r holds
    // rows r (lanes 0-15) and r+8 (lanes 16-31), N = lane within tile.
    {
      const int j = wave * 16 + subl;
#pragma unroll
      for (int r = 0; r < 8; ++r) {
        const int m = r + 8 * half;
        sh_upd[m * APAD + j] = sh_mrow[m] + __logf(acc[r]);
      }
    }
    __syncthreads();

    // Phase D: alpha = mask ? upd + emission : alpha  (coalesced global read)
#pragma unroll
    for (int u = 0; u < (BT * TT) / 256; ++u) {
      const int e = tid + 256 * u;
      const int m = e >> 7, j = e & 127;
      const bool mk = mask[(size_t)(b0 + m) * SS + s] != 0;
      const float nv =
          sh_upd[m * APAD + j] + inputs[((size_t)(b0 + m) * SS + s) * TT + j];
      if (mk) sh_alpha[m * TT + j] = nv;
    }
    // Prefetch next step's emission tile (global_prefetch_b8).
    if (s + 1 < SS) {
      __builtin_prefetch(
          &inputs[((size_t)(b0 + (tid >> 4)) * SS + s + 1) * TT + ((tid & 15) << 3)],
          0, 1);
    }
    __syncthreads();
  }

  // Final: log_den[row] = logsumexp_j(alpha[row][j] + end[j]); accumulate -log_den.
  {
    const int row = 2 * wave + half;
    float mx = -3.4e38f;
    for (int c = subl; c < TT; c += 16)
      mx = fmaxf(mx, sh_alpha[row * TT + c] + endt[c]);
    mx = fmaxf(mx, __shfl_xor(mx, 1));
    mx = fmaxf(mx, __shfl_xor(mx, 2));
    mx = fmaxf(mx, __shfl_xor(mx, 4));
    mx = fmaxf(mx, __shfl_xor(mx, 8));
    float sum = 0.0f;
    for (int c = subl; c < TT; c += 16)
      sum += __expf(sh_alpha[row * TT + c] + endt[c] - mx);
    sum += __shfl_xor(sum, 1);
    sum += __shfl_xor(sum, 2);
    sum += __shfl_xor(sum, 4);
    sum += __shfl_xor(sum, 8);
    if (subl == 0) atomicAdd(out, -(mx + __logf(sum)));
  }
}

extern "C" void kernel_launch(void* const* d_in, const int* in_sizes, int n_in,
                              void* d_out, int out_size, void* d_ws, size_t ws_size,
                              hipStream_t stream) {
  const float* inputs = (const float*)d_in[0];          // (B,S,T) f32
  const int* tags = (const int*)d_in[1];                // (B,S) i32
  const unsigned char* mask = (const unsigned char*)d_in[2];  // (B,S) bool
  const float* trans = (const float*)d_in[3];           // (T,T) f32
  const float* startt = (const float*)d_in[4];          // (T,) f32
  const float* endt = (const float*)d_in[5];            // (T,) f32
  float* out = (float*)d_out;                           // scalar f32

  crf_zero_out<<<1, 1, 0, stream>>>(out);
  crf_num_kernel<<<BB, 128, 0, stream>>>(inputs, tags, mask, trans, startt, endt, out);
  crf_den_kernel<<<BB / BT, 256, 0, stream>>>(inputs, mask, trans, startt, endt, out);
}